// ConformalGCN_42468636623302
// MI455X (gfx1250) — compile-verified
//
#include <hip/hip_runtime.h>

typedef __attribute__((ext_vector_type(2))) float v2f;
typedef __attribute__((ext_vector_type(8))) float v8f;

#define TPB 256

// ---------------- degree / norm ----------------
__global__ void deg_init_k(float* deg, int n) {
    int i = blockIdx.x * blockDim.x + threadIdx.x;
    if (i < n) deg[i] = 1.0f;   // self loop contributes 1
}

__global__ void deg_count_k(const int* __restrict__ dst, float* deg, int e) {
    int i = blockIdx.x * blockDim.x + threadIdx.x;
    if (i < e) unsafeAtomicAdd(&deg[dst[i]], 1.0f);
}

__global__ void deg_to_dinv_k(float* deg, int n) {
    int i = blockIdx.x * blockDim.x + threadIdx.x;
    if (i < n) deg[i] = rsqrtf(deg[i]);   // deg >= 1 always (self loops)
}

// ---------------- f32 WMMA GEMM: C[MxNc] = A[MxK] @ B[KxNc] ----------------
// M % 16 == 0, K % 4 == 0, Nc % 16 == 0. One 16x16 output tile per wave.
__global__ void gemm_f32_wmma_k(const float* __restrict__ A, const float* __restrict__ B,
                                float* __restrict__ C, int M, int K, int Nc) {
    int wave  = (blockIdx.x * blockDim.x + threadIdx.x) >> 5;
    int lane  = threadIdx.x & 31;
    int ntile = Nc >> 4;
    int tileN = wave % ntile;
    int tileM = wave / ntile;
    if (tileM * 16 >= M) return;            // wave-uniform: EXEC stays all-1s

    int half = lane >> 4;                   // 0: K pair {0,1}; 1: K pair {2,3}
    int l15  = lane & 15;
    int rowA = tileM * 16 + l15;            // A: lane holds row M=l15
    int colB = tileN * 16 + l15;            // B: lane holds col N=l15

    v8f acc = {};
    const float* Arow = A + (size_t)rowA * K;
    for (int k0 = 0; k0 < K; k0 += 4) {
        int ka = k0 + half * 2;
        v2f a, b;
        a.x = Arow[ka];
        a.y = Arow[ka + 1];
        b.x = B[(size_t)ka * Nc + colB];
        b.y = B[(size_t)(ka + 1) * Nc + colB];
        acc = __builtin_amdgcn_wmma_f32_16x16x4_f32(
            /*neg_a=*/false, a, /*neg_b=*/false, b,
            /*c_mod=*/(short)0, acc, /*reuse_a=*/false, /*reuse_b=*/false);
    }
    // C/D layout: VGPR r -> M = r (lanes 0-15) or M = r+8 (lanes 16-31), N = l15
    int rowBase = tileM * 16 + 8 * half;
#pragma unroll
    for (int r = 0; r < 8; ++r)
        C[(size_t)(rowBase + r) * Nc + colB] = acc[r];
}

// ---------------- self-loop init: agg[i][c] = h[i][c] * dinv[i]^2 ----------------
__global__ void self_init_k(const float* __restrict__ h, const float* __restrict__ dinv,
                            float* __restrict__ agg, int n, int F) {
    int idx = blockIdx.x * blockDim.x + threadIdx.x;
    int total = n * F;
    if (idx >= total) return;
    int node = idx / F;
    float di = dinv[node];
    agg[idx] = h[idx] * di * di;
}

// ---------------- edge scatter: agg[dst] += h[src] * dinv[src]*dinv[dst] ----------
// One thread per (edge, group of 4 channels); F4 = F/4 groups per edge.
__global__ void scatter_edges_k(const float* __restrict__ h,
                                const int* __restrict__ src, const int* __restrict__ dst,
                                const float* __restrict__ dinv,
                                float* __restrict__ agg, int e, int F4, int F) {
    int idx = blockIdx.x * blockDim.x + threadIdx.x;
    int total = e * F4;
    if (idx >= total) return;
    int ed = idx / F4;
    int g  = idx - ed * F4;
    int s = src[ed];
    int d = dst[ed];
    float nrm = dinv[s] * dinv[d];
    const float4 v = *((const float4*)(h + (size_t)s * F) + g);
    float* ap = agg + (size_t)d * F + g * 4;
    unsafeAtomicAdd(ap + 0, v.x * nrm);
    unsafeAtomicAdd(ap + 1, v.y * nrm);
    unsafeAtomicAdd(ap + 2, v.z * nrm);
    unsafeAtomicAdd(ap + 3, v.w * nrm);
}

// ---------------- bias (+ optional relu) ----------------
__global__ void bias_relu_k(float* a, const float* __restrict__ b, int total, int F) {
    int idx = blockIdx.x * blockDim.x + threadIdx.x;
    if (idx >= total) return;
    int c = idx % F;
    a[idx] = fmaxf(a[idx] + b[c], 0.0f);
}

__global__ void bias_add_k(float* a, const float* __restrict__ b, int total, int F) {
    int idx = blockIdx.x * blockDim.x + threadIdx.x;
    if (idx >= total) return;
    int c = idx % F;
    a[idx] = a[idx] + b[c];
}

static inline int cdiv(long long a, int b) { return (int)((a + b - 1) / b); }

extern "C" void kernel_launch(void* const* d_in, const int* in_sizes, int n_in,
                              void* d_out, int out_size, void* d_ws, size_t ws_size,
                              hipStream_t stream) {
    const float* x  = (const float*)d_in[0];
    const int*   ei = (const int*)d_in[1];     // [2, E]: row 0 = src, row 1 = dst
    const float* W1 = (const float*)d_in[2];
    const float* b1 = (const float*)d_in[3];
    const float* W2 = (const float*)d_in[4];
    const float* b2 = (const float*)d_in[5];
    float* out = (float*)d_out;

    const int IN  = 128;
    const int HID = in_sizes[3];               // 64
    const int OUT = in_sizes[5];               // 96
    const int N   = in_sizes[0] / IN;          // 100000
    const int E   = in_sizes[1] / 2;           // 1600000

    const int* src = ei;
    const int* dst = ei + E;

    // workspace: dinv[N] | bufA[N*OUT] (h, then h2) | bufB[N*HID] (agg1 / h_relu)
    float* dinv = (float*)d_ws;
    float* bufA = dinv + N;
    float* bufB = bufA + (size_t)N * OUT;

    // 1) degree (with self loop) -> dinv
    deg_init_k   <<<cdiv(N, TPB), TPB, 0, stream>>>(dinv, N);
    deg_count_k  <<<cdiv(E, TPB), TPB, 0, stream>>>(dst, dinv, E);
    deg_to_dinv_k<<<cdiv(N, TPB), TPB, 0, stream>>>(dinv, N);

    // 2) h = x @ W1   (WMMA f32)
    {
        int waves = (N / 16) * (HID / 16);
        gemm_f32_wmma_k<<<cdiv((long long)waves * 32, TPB), TPB, 0, stream>>>(x, W1, bufA, N, IN, HID);
    }

    // 3) agg1 = self-loop + edge scatter
    self_init_k<<<cdiv((long long)N * HID, TPB), TPB, 0, stream>>>(bufA, dinv, bufB, N, HID);
    scatter_edges_k<<<cdiv((long long)E * (HID / 4), TPB), TPB, 0, stream>>>(
        bufA, src, dst, dinv, bufB, E, HID / 4, HID);

    // 4) h_relu = relu(agg1 + b1)   (in place in bufB)
    bias_relu_k<<<cdiv((long long)N * HID, TPB), TPB, 0, stream>>>(bufB, b1, N * HID, HID);

    // 5) h2 = h_relu @ W2   (WMMA f32, reuse bufA)
    {
        int waves = (N / 16) * (OUT / 16);
        gemm_f32_wmma_k<<<cdiv((long long)waves * 32, TPB), TPB, 0, stream>>>(bufB, W2, bufA, N, HID, OUT);
    }

    // 6) out = self-loop + edge scatter + b2
    self_init_k<<<cdiv((long long)N * OUT, TPB), TPB, 0, stream>>>(bufA, dinv, out, N, OUT);
    scatter_edges_k<<<cdiv((long long)E * (OUT / 4), TPB), TPB, 0, stream>>>(
        bufA, src, dst, dinv, out, E, OUT / 4, OUT);
    bias_add_k<<<cdiv((long long)N * OUT, TPB), TPB, 0, stream>>>(out, b2, N * OUT, OUT);
}